// STVisualTokenSelection_79980880986200
// MI455X (gfx1250) — compile-verified
//
#include <hip/hip_runtime.h>

// Shapes (from reference setup): B=8, F=12 -> BF=96, L=196, D=1024, NS=256
#define BF_    96
#define L_     196
#define D_     1024
#define NS_    256
#define KSEL   49       // L/4
#define LNK    147      // L - KSEL
#define NMASK  73       // (L-KSEL)/2
#define INV_NS (1.0f/256.0f)
#define ASTRIDE 198     // LDS row stride for A tile (even -> 8B-aligned v2f loads)

typedef __attribute__((ext_vector_type(2))) float v2f;
typedef __attribute__((ext_vector_type(8))) float v8f;

// ---------------------------------------------------------------- zero scratch
__global__ void zero_f32(float* __restrict__ p, int n) {
    int i = blockIdx.x * blockDim.x + threadIdx.x;
    if (i < n) p[i] = 0.0f;
}

// --------------------------------------------------- per-(b,s) top-k indicator
// One wave per (b,s) row. score==1 exactly (softmax over size-1 axis), so
// pert = 1 + noise*0.05 computed with the reference's f32 rounding order.
__global__ __launch_bounds__(128) void topk_indicator(const float* __restrict__ noise,
                                                      float* __restrict__ ind) {
    __shared__ float vals[4][L_];
    __shared__ int   flg[4][L_];
    const int w    = threadIdx.x >> 5;
    const int lane = threadIdx.x & 31;
    const int rid  = blockIdx.x * 4 + w;         // 0 .. BF*NS-1
    const int b    = rid / NS_;
    const float* row = noise + (size_t)rid * L_;

    for (int l = lane; l < L_; l += 32)
        vals[w][l] = __fadd_rn(1.0f, __fmul_rn(row[l], 0.05f));
    __syncthreads();

    for (int l = lane; l < L_; l += 32) {
        float v = vals[w][l];
        int rank = 0;
        for (int m = 0; m < L_; ++m) {
            float u = vals[w][m];
            rank += (u > v) || ((u == v) && (m < l));   // jax top_k tie-break
        }
        flg[w][l] = (rank < KSEL) ? 1 : 0;
    }
    __syncthreads();

    for (int l = lane; l < L_; l += 32) {
        int p = 0;
        for (int m = 0; m < l; ++m) p += flg[w][m];     // #topk indices < l
        int r = flg[w][l] ? p : (KSEL + (l - p));       // row in combined indicator
        atomicAdd(&ind[((size_t)b * L_ + r) * L_ + l], INV_NS);
    }
}

// ------------------------------------------------------------------- mask rows
// mask[b,j] = 1 iff mask_rand[b,j] is among the NMASK smallest (tie: low index)
__global__ __launch_bounds__(32) void mask_kernel(const float* __restrict__ mr,
                                                  float* __restrict__ mask) {
    __shared__ float vals[LNK];
    const int b = blockIdx.x;
    const int lane = threadIdx.x;
    const float* row = mr + (size_t)b * LNK;
    for (int j = lane; j < LNK; j += 32) vals[j] = row[j];
    __syncthreads();
    for (int j = lane; j < LNK; j += 32) {
        float v = vals[j];
        int rank = 0;
        for (int m = 0; m < LNK; ++m) {
            float u = vals[m];
            rank += (u < v) || ((u == v) && (m < j));
        }
        mask[(size_t)b * LNK + j] = (rank < NMASK) ? 1.0f : 0.0f;
    }
}

// ------------------------------------------------- batched indicator GEMM, f32
// Per b: (196x196 indicator) @ (196x1024 x[b]) via V_WMMA_F32_16X16X4_F32.
// grid = (96, 13 Mtiles, 8), 8 waves/WG -> 64 N-tiles; A tile staged in LDS.
__global__ __launch_bounds__(256) void ind_gemm(const float* __restrict__ ind,
                                                const float* __restrict__ x,
                                                const float* __restrict__ mask,
                                                float* __restrict__ out) {
    __shared__ float As[16 * ASTRIDE];
    const int b      = blockIdx.x;
    const int m_base = blockIdx.y * 16;
    const int tid    = threadIdx.x;

    const float* indb = ind + (size_t)b * L_ * L_;
    for (int i = tid; i < 16 * L_; i += 256) {
        int r = i / L_, k = i - r * L_;
        int m = m_base + r;
        As[r * ASTRIDE + k] = (m < L_) ? indb[(size_t)m * L_ + k] : 0.0f; // zero-pad rows
    }
    __syncthreads();

    const int wave  = tid >> 5;
    const int lane  = tid & 31;
    const int l16   = lane & 15;
    const int hi    = lane >> 4;                  // 0 or 1
    const int nbase = (blockIdx.z * 8 + wave) * 16;
    const float* xb = x + (size_t)b * L_ * D_;

    // A fragment: lane holds A[l16][k0+2*hi], A[l16][k0+2*hi+1]
    const float* arow = As + l16 * ASTRIDE + 2 * hi;
    // B fragment: lane holds x[k0+2*hi][nbase+l16], x[k0+2*hi+1][nbase+l16]
    const float* bcol = xb + nbase + l16 + (size_t)(2 * hi) * D_;

    v8f acc = {};
    #pragma unroll 7
    for (int k0 = 0; k0 < L_; k0 += 4) {          // 49 K-steps of 4
        v2f a = *(const v2f*)(arow + k0);
        v2f bb;
        bb.x = bcol[(size_t)k0 * D_];
        bb.y = bcol[(size_t)k0 * D_ + D_];
        acc = __builtin_amdgcn_wmma_f32_16x16x4_f32(
                /*neg_a=*/false, a, /*neg_b=*/false, bb,
                /*c_mod=*/(short)0, acc, /*reuse_a=*/false, /*reuse_b=*/false);
    }

    // D layout: VGPR r -> row m_base + r + 8*hi, col nbase + l16
    float* out_sel = out;
    float* out_non = out + (size_t)BF_ * KSEL * D_;
    float* out_msk = out_non + (size_t)BF_ * LNK * D_;
    const int col = nbase + l16;
    #pragma unroll
    for (int r = 0; r < 8; ++r) {
        int m = m_base + r + 8 * hi;
        float v = acc[r];
        if (m < KSEL) {
            out_sel[((size_t)b * KSEL + m) * D_ + col] = v;
        } else if (m < L_) {
            int j = m - KSEL;
            size_t o = ((size_t)b * LNK + j) * D_ + col;
            out_non[o] = v;
            out_msk[o] = v * mask[(size_t)b * LNK + j];
        }
    }
}

// ------------------------------------------------------------------- launcher
extern "C" void kernel_launch(void* const* d_in, const int* in_sizes, int n_in,
                              void* d_out, int out_size, void* d_ws, size_t ws_size,
                              hipStream_t stream) {
    const float* x     = (const float*)d_in[0];   // (96,196,1024)
    const float* noise = (const float*)d_in[1];   // (96,256,196)
    const float* mrand = (const float*)d_in[2];   // (96,147)
    // d_in[3..7] (ln_gamma, ln_beta, W_in, W_out1, W_out2) are mathematically
    // dead: softmax over a size-1 axis makes score identically 1.0, and a
    // constant shift cannot change top_k membership.

    float* ind  = (float*)d_ws;                       // 96*196*196 f32 = 14.75 MB
    float* mask = ind + (size_t)BF_ * L_ * L_;        // 96*147 f32
    float* out  = (float*)d_out;

    const int n_ind = BF_ * L_ * L_;
    zero_f32<<<(n_ind + 255) / 256, 256, 0, stream>>>(ind, n_ind);
    topk_indicator<<<BF_ * NS_ / 4, 128, 0, stream>>>(noise, ind);
    mask_kernel<<<BF_, 32, 0, stream>>>(mrand, mask);

    dim3 g(BF_, 13, 8);   // b, M-tiles (208 rows padded), N-tile groups of 8
    ind_gemm<<<g, 256, 0, stream>>>(ind, x, mask, out);
}